// SSIM_66065186947315
// MI455X (gfx1250) — compile-verified
//
#include <hip/hip_runtime.h>
#include <hip/hip_bf16.h>

typedef __attribute__((ext_vector_type(2))) float v2f;
typedef __attribute__((ext_vector_type(8))) float v8f;

#define IMG_H 512
#define IMG_W 512
#define PLANES 48          // B*C = 16*3
#define TILES_X 32         // 512/16
#define TILES_PER_PLANE (TILES_X * TILES_X)
#define TOTAL_TILES (PLANES * TILES_PER_PLANE)   // 49152
#define WPB 4              // waves (tiles) per block
#define WAVE_LDS_F 3808    // 784 (x) + 784 (y) + 5*448 (T) floats per wave

#define SSIM_C1 0.0001f
#define SSIM_C2 0.0009f

// Normalized 1D Gaussian, sigma = 1.5, window 11 (matches reference)
__device__ const float c_W11[11] = {
    0.00102838f, 0.00759875f, 0.03600077f, 0.10936069f, 0.21300553f,
    0.26601172f, 0.21300553f, 0.10936069f, 0.03600077f, 0.00759875f,
    0.00102838f};

__device__ __forceinline__ float wsel(int d) {
  // w[d] if 0 <= d <= 10 else 0  (band-matrix element)
  return ((unsigned)d <= 10u) ? c_W11[d] : 0.0f;
}

__device__ __forceinline__ v8f wmma4(v2f a, v2f b, v8f c) {
  // D(16x16,f32) = A(16x4,f32) * B(4x16,f32) + C
  return __builtin_amdgcn_wmma_f32_16x16x4_f32(false, a, false, b, (short)0, c,
                                               false, false);
}

__device__ __forceinline__ void wait_lds() {
  // All LDS producer/consumer pairs are intra-wave (DS ops are in-order per
  // wave); this is a belt-and-braces drain + compiler reordering fence.
  asm volatile("s_wait_dscnt 0x0" ::: "memory");
}

__launch_bounds__(WPB * 32)
__global__ void ssim_wmma_kernel(const float* __restrict__ img1,
                                 const float* __restrict__ img2,
                                 float* __restrict__ out) {
  __shared__ float lds[WPB * WAVE_LDS_F];

  const int wave = threadIdx.x >> 5;
  const int lane = threadIdx.x & 31;
  const int l    = lane & 15;        // M (A) / N (B,C,D) index within half
  const int half = lane >> 4;        // 0: K pair {0,1}, 1: K pair {2,3}
  const int kh   = half << 1;

  const int tileId = blockIdx.x * WPB + wave;
  const int plane  = tileId >> 10;                 // / TILES_PER_PLANE
  const int t      = tileId & (TILES_PER_PLANE - 1);
  const int tr     = (t >> 5) << 4;                // tile row origin
  const int tc     = (t & 31) << 4;                // tile col origin

  float* xb = lds + wave * WAVE_LDS_F;   // 28 x 28 halo region of img1
  float* yb = xb + 784;                  // 28 x 28 halo region of img2
  float* tb = yb + 784;                  // 5 channels x 16 x 28 intermediates

  // ---- Phase 1: global -> LDS (zero padding outside image, like JAX conv) --
  const size_t pbase = (size_t)plane * (IMG_H * IMG_W);
  for (int i = lane; i < 28 * 28; i += 32) {
    const int r = i / 28, c = i % 28;
    const int gr = tr - 5 + r, gc = tc - 5 + c;
    const bool ok = (gr >= 0) & (gr < IMG_H) & (gc >= 0) & (gc < IMG_W);
    float xv = 0.0f, yv = 0.0f;
    if (ok) {
      const size_t gidx = pbase + (size_t)gr * IMG_W + gc;
      xv = img1[gidx];
      yv = img2[gidx];
    }
    xb[i] = xv;
    yb[i] = yv;
  }
  wait_lds();

  // Band-matrix fragments: Wv[m][j] = w[j-m] (A of vertical pass) and
  // Wh[j][n] = w[j-n] (B of horizontal pass) produce identical per-lane data.
  v2f wf[7];
#pragma unroll
  for (int s = 0; s < 7; ++s) {
    const int j0 = 4 * s + kh;
    wf[s].x = wsel(j0 - l);
    wf[s].y = wsel(j0 + 1 - l);
  }

  // ---- Phase 2: vertical pass, D = Wv * X, 2 col-groups x 5 channels ------
  v8f vx[2], vy[2], vxx[2], vyy[2], vxy[2];
#pragma unroll
  for (int cg = 0; cg < 2; ++cg) {
    vx[cg] = (v8f){}; vy[cg] = (v8f){}; vxx[cg] = (v8f){};
    vyy[cg] = (v8f){}; vxy[cg] = (v8f){};
  }
#pragma unroll
  for (int s = 0; s < 7; ++s) {
    const int j0 = 4 * s + kh;           // input row pair for this lane-half
#pragma unroll
    for (int cg = 0; cg < 2; ++cg) {
      const int col = (cg << 4) + l;
      const float x0 = xb[j0 * 28 + col];
      const float x1 = xb[(j0 + 1) * 28 + col];
      const float y0 = yb[j0 * 28 + col];
      const float y1 = yb[(j0 + 1) * 28 + col];
      vx[cg]  = wmma4(wf[s], (v2f){x0, x1},           vx[cg]);
      vy[cg]  = wmma4(wf[s], (v2f){y0, y1},           vy[cg]);
      vxx[cg] = wmma4(wf[s], (v2f){x0 * x0, x1 * x1}, vxx[cg]);
      vyy[cg] = wmma4(wf[s], (v2f){y0 * y0, y1 * y1}, vyy[cg]);
      vxy[cg] = wmma4(wf[s], (v2f){x0 * y0, x1 * y1}, vxy[cg]);
    }
  }

  // ---- Phase 3: accumulators -> LDS (transpose to A-fragment layout) ------
#pragma unroll
  for (int cg = 0; cg < 2; ++cg) {
    const int n = (cg << 4) + l;
#pragma unroll
    for (int v = 0; v < 8; ++v) {
      const int m = v + (half << 3);
      if (n < 26) {
        tb[0 * 448 + m * 28 + n] = vx[cg][v];
        tb[1 * 448 + m * 28 + n] = vy[cg][v];
        tb[2 * 448 + m * 28 + n] = vxx[cg][v];
        tb[3 * 448 + m * 28 + n] = vyy[cg][v];
        tb[4 * 448 + m * 28 + n] = vxy[cg][v];
      } else if (n < 28) {
        // K-padding columns must be finite (weights there are zero)
        tb[0 * 448 + m * 28 + n] = 0.0f;
        tb[1 * 448 + m * 28 + n] = 0.0f;
        tb[2 * 448 + m * 28 + n] = 0.0f;
        tb[3 * 448 + m * 28 + n] = 0.0f;
        tb[4 * 448 + m * 28 + n] = 0.0f;
      }
    }
  }
  wait_lds();

  // ---- Phase 4: horizontal pass, D = T * Wh ------------------------------
  v8f hx = (v8f){}, hy = (v8f){}, hxx = (v8f){}, hyy = (v8f){}, hxy = (v8f){};
#pragma unroll
  for (int s = 0; s < 7; ++s) {
    const int j0 = 4 * s + kh;
    const int row = l * 28 + j0;
    hx  = wmma4((v2f){tb[0 * 448 + row], tb[0 * 448 + row + 1]}, wf[s], hx);
    hy  = wmma4((v2f){tb[1 * 448 + row], tb[1 * 448 + row + 1]}, wf[s], hy);
    hxx = wmma4((v2f){tb[2 * 448 + row], tb[2 * 448 + row + 1]}, wf[s], hxx);
    hyy = wmma4((v2f){tb[3 * 448 + row], tb[3 * 448 + row + 1]}, wf[s], hyy);
    hxy = wmma4((v2f){tb[4 * 448 + row], tb[4 * 448 + row + 1]}, wf[s], hxy);
  }

  // ---- Phase 5: SSIM map + store -----------------------------------------
#pragma unroll
  for (int v = 0; v < 8; ++v) {
    const int m = v + (half << 3);
    const float mu1 = hx[v], mu2 = hy[v];
    const float mu1sq = mu1 * mu1, mu2sq = mu2 * mu2, mu12 = mu1 * mu2;
    const float s1 = hxx[v] - mu1sq;
    const float s2 = hyy[v] - mu2sq;
    const float s12 = hxy[v] - mu12;
    const float num = (2.0f * mu12 + SSIM_C1) * (2.0f * s12 + SSIM_C2);
    const float den = (mu1sq + mu2sq + SSIM_C1) * (s1 + s2 + SSIM_C2);
    out[pbase + (size_t)(tr + m) * IMG_W + (tc + l)] = num / den;
  }
}

extern "C" void kernel_launch(void* const* d_in, const int* in_sizes, int n_in,
                              void* d_out, int out_size, void* d_ws,
                              size_t ws_size, hipStream_t stream) {
  (void)in_sizes; (void)n_in; (void)d_ws; (void)ws_size; (void)out_size;
  const float* img1 = (const float*)d_in[0];
  const float* img2 = (const float*)d_in[1];
  // d_in[2] is window_size == 11, baked into the kernel.
  float* out = (float*)d_out;
  dim3 grid(TOTAL_TILES / WPB);   // 12288
  dim3 block(WPB * 32);           // 128 threads = 4 waves, 1 tile per wave
  hipLaunchKernelGGL(ssim_wmma_kernel, grid, block, 0, stream, img1, img2, out);
}